// ToDenseBEVConvolution_8529805050238
// MI455X (gfx1250) — compile-verified
//
#include <hip/hip_runtime.h>

typedef __attribute__((ext_vector_type(2))) float v2f;
typedef __attribute__((ext_vector_type(8))) float v8f;

#define H_DIM 256
#define W_DIM 256
#define NBINS 16
#define CIN   64
#define COUT  64
#define HW    (H_DIM * W_DIM)        /* 65536 */
#define PLANE (COUT * HW)            /* 4194304 per batch image */

// ---------------- counting sort by height bin ----------------

__global__ void __launch_bounds__(256)
hist_kernel(const int* __restrict__ coords, const int* __restrict__ stride_p,
            int n, int* __restrict__ counts) {
    int i = blockIdx.x * blockDim.x + threadIdx.x;
    if (i < n) {
        int s = *stride_p;
        int k = coords[i * 4 + 1] / s;
        atomicAdd(&counts[k], 1);
    }
}

__global__ void scan_kernel(const int* __restrict__ counts,
                            int* __restrict__ offsets,
                            int* __restrict__ cursors) {
    if (threadIdx.x == 0) {
        int acc = 0;
        for (int i = 0; i < NBINS; ++i) {
            offsets[i] = acc;
            cursors[i] = acc;
            acc += counts[i];
        }
        offsets[NBINS] = acc;
    }
}

__global__ void __launch_bounds__(256)
order_kernel(const int* __restrict__ coords, const int* __restrict__ stride_p,
             int n, int* __restrict__ cursors, int* __restrict__ order) {
    int i = blockIdx.x * blockDim.x + threadIdx.x;
    if (i < n) {
        int s = *stride_p;
        int k = coords[i * 4 + 1] / s;
        int pos = atomicAdd(&cursors[k], 1);
        order[pos] = i;
    }
}

// ---------------- 16-point tile GEMM + scatter-add ----------------
// One wave32 per block. A = 16x64 feats tile (LDS), B = 64x64 bin kernel
// (global, L2-hot), D = 16x64 out tile, scattered with global_atomic_add_f32.

__device__ __forceinline__ void fadd_atomic(float* p, float v) {
    unsafeAtomicAdd(p, v);   // hardware global_atomic_add_f32 (no return)
}

__global__ void __launch_bounds__(32)
bev_wmma_kernel(const int* __restrict__ coords,
                const float* __restrict__ feats,
                const float* __restrict__ kern,
                const int* __restrict__ stride_p,
                const int* __restrict__ counts,
                const int* __restrict__ offsets,
                const int* __restrict__ order,
                float* __restrict__ out) {
    const int bin   = blockIdx.y;
    const int cnt   = counts[bin];
    const int tile0 = blockIdx.x * 16;
    if (tile0 >= cnt) return;                 // uniform early-exit (before WMMA)

    const int start = offsets[bin];
    const int lane  = threadIdx.x;
    const int s     = *stride_p;

    __shared__ int   s_pt[16];
    __shared__ int   s_base[16];
    __shared__ float Asm[16][68];             // stride 68 -> conflict-free frag reads

    // per-row point index + flat BEV base (b*PLANE + x*W + z); -1 = padding row
    if (lane < 16) {
        int idx  = tile0 + lane;
        int p    = (idx < cnt) ? order[start + idx] : -1;
        int base = -1;
        if (p >= 0) {
            int4 c = ((const int4*)coords)[p];           // x,y,z,b
            base = c.w * PLANE + (c.x / s) * W_DIM + (c.z / s);
        }
        s_pt[lane]   = p;
        s_base[lane] = base;
    }
    __syncthreads();

    // stage A tile: 16 rows x 64 floats, coalesced float4 loads; pad rows -> 0
    for (int i = lane; i < 16 * 16; i += 32) {
        int r  = i >> 4;
        int c4 = i & 15;
        float4 v = make_float4(0.f, 0.f, 0.f, 0.f);
        int p = s_pt[r];
        if (p >= 0) v = ((const float4*)feats)[p * 16 + c4];
        float* row = &Asm[r][c4 * 4];
        row[0] = v.x; row[1] = v.y; row[2] = v.z; row[3] = v.w;
    }
    __syncthreads();

    const float* __restrict__ B = kern + bin * (CIN * COUT);
    const int m    = lane & 15;          // A row / D column-within-halfwave
    const int koff = (lane >> 4) * 2;    // upper halfwave holds K+2,K+3
    const int ncol = lane & 15;          // B/D column within 16-wide n-tile

    v8f acc0 = {}, acc1 = {}, acc2 = {}, acc3 = {};

#pragma unroll
    for (int kk = 0; kk < 16; ++kk) {
        const int kb = kk * 4 + koff;
        v2f a;
        a.x = Asm[m][kb];
        a.y = Asm[m][kb + 1];
        const float* brow = B + kb * COUT + ncol;
        v2f b0; b0.x = brow[0];  b0.y = brow[COUT + 0];
        v2f b1; b1.x = brow[16]; b1.y = brow[COUT + 16];
        v2f b2; b2.x = brow[32]; b2.y = brow[COUT + 32];
        v2f b3; b3.x = brow[48]; b3.y = brow[COUT + 48];
        acc0 = __builtin_amdgcn_wmma_f32_16x16x4_f32(false, a, false, b0, (short)0, acc0, false, false);
        acc1 = __builtin_amdgcn_wmma_f32_16x16x4_f32(false, a, false, b1, (short)0, acc1, false, false);
        acc2 = __builtin_amdgcn_wmma_f32_16x16x4_f32(false, a, false, b2, (short)0, acc2, false, false);
        acc3 = __builtin_amdgcn_wmma_f32_16x16x4_f32(false, a, false, b3, (short)0, acc3, false, false);
    }

    // D layout: VGPR j holds row j (lanes 0-15) / row j+8 (lanes 16-31), col = ncol
    const int mtop = (lane >> 4) << 3;   // 0 or 8
#pragma unroll
    for (int j = 0; j < 8; ++j) {
        int mm   = j + mtop;
        int base = s_base[mm];
        if (base < 0) continue;          // padding row -> skip scatter
        fadd_atomic(&out[base + (ncol +  0) * HW], acc0[j]);
        fadd_atomic(&out[base + (ncol + 16) * HW], acc1[j]);
        fadd_atomic(&out[base + (ncol + 32) * HW], acc2[j]);
        fadd_atomic(&out[base + (ncol + 48) * HW], acc3[j]);
    }
}

// ---------------- launch ----------------

extern "C" void kernel_launch(void* const* d_in, const int* in_sizes, int n_in,
                              void* d_out, int out_size, void* d_ws, size_t ws_size,
                              hipStream_t stream) {
    const int*   coords   = (const int*)d_in[0];
    const float* feats    = (const float*)d_in[1];
    const float* kern     = (const float*)d_in[2];
    const int*   stride_p = (const int*)d_in[3];
    float*       out      = (float*)d_out;

    const int n = in_sizes[0] / 4;       // coords is [N,4]

    int* ws      = (int*)d_ws;
    int* counts  = ws;                   // 16
    int* offsets = ws + 16;              // 17
    int* cursors = ws + 40;              // 16
    int* order   = ws + 64;              // n

    // zero output grid (scatter target) and the sort counters
    hipMemsetAsync(d_out, 0, (size_t)out_size * sizeof(float), stream);
    hipMemsetAsync(d_ws, 0, 64 * sizeof(int), stream);

    const int tb = 256;
    hist_kernel <<<(n + tb - 1) / tb, tb, 0, stream>>>(coords, stride_p, n, counts);
    scan_kernel <<<1, 32, 0, stream>>>(counts, offsets, cursors);
    order_kernel<<<(n + tb - 1) / tb, tb, 0, stream>>>(coords, stride_p, n, cursors, order);

    const int tiles = (n + 15) / 16;     // per-bin upper bound; empty tiles exit early
    dim3 grid(tiles, NBINS);
    bev_wmma_kernel<<<grid, 32, 0, stream>>>(coords, feats, kern, stride_p,
                                             counts, offsets, order, out);
}